// HGIB_68650757259657
// MI455X (gfx1250) — compile-verified
//
#include <hip/hip_runtime.h>
#include <math.h>
#include <stdint.h>

// ---------------------------------------------------------------------------
// HGIB-style multi-behavior GNN forward for MI455X (gfx1250, wave32).
//  - Edge dot-products: V_WMMA_F32_16X16X4_F32, 16 edges per wave, K-loop x16.
//  - Segment-sum conv: float4 gathers (tables fit in 192MB L2) + HW f32 atomics.
//  - Row normalize: 16 lanes/row, shfl_xor reduction (wave32 native).
// ---------------------------------------------------------------------------

#define N_USERS_C 100000
#define N_ITEMS_C 100000
#define DIM       64
#define NEDGE     1000000
#define NTOT      (N_USERS_C + N_ITEMS_C + 2)   // 200002
#define THRESH    0.05f

typedef float v2f __attribute__((ext_vector_type(2)));
typedef float v8f __attribute__((ext_vector_type(8)));

// --- deterministic per-edge hash -> uniform(0,1) ---------------------------
__device__ __forceinline__ float hash_u01(uint32_t x) {
    x ^= x >> 17; x *= 0xED5AD4BBu;
    x ^= x >> 11; x *= 0xAC4C1B51u;
    x ^= x >> 15; x *= 0x31848BABu;
    x ^= x >> 14;
    return ((float)(x >> 8) + 0.5f) * (1.0f / 16777216.0f);  // (0,1)
}

// ---------------------------------------------------------------------------
// Kernel 1: edge gather-dot logits + hard gumbel-sigmoid weights.
// One wave handles 16 edges via WMMA f32 16x16x4:
//   A[m][k] = X[src[m]][k0+k]   (16x4 f32, 2 VGPRs: lane m(+16), {k0+2kh, +1})
//   B[k][n] = X[dst[n]][k0+k]   (4x16 f32, mirrored lane layout)
//   accumulate D = A*B + D over k0 = 0..60 step 4; diagonal = edge dots.
// EXEC is all-ones for every wave that reaches the WMMA (E % 16 == 0).
// ---------------------------------------------------------------------------
__global__ void edge_gumbel_wmma_kernel(const float* __restrict__ X,
                                        const int*   __restrict__ e0,
                                        const int*   __restrict__ e1,
                                        float*       __restrict__ w_out,
                                        int E, uint32_t seed)
{
    const int lane   = threadIdx.x & 31;
    const int waveIb = threadIdx.x >> 5;
    const long wave  = (long)blockIdx.x * (blockDim.x >> 5) + waveIb;
    const long tile  = wave * 16;
    if (tile >= E) return;                       // wave-uniform guard

    const int mlo = lane & 15;                   // edge slot within tile
    const int kh  = lane >> 4;                   // K half (0 or 1)

    const int src = e0[tile + mlo];
    const int dst = e1[tile + mlo];
    const float* rowA = X + (size_t)src * DIM;
    const float* rowB = X + (size_t)dst * DIM;

    v8f c = {};
#pragma unroll
    for (int k0 = 0; k0 < DIM; k0 += 4) {
        v2f a = *(const v2f*)(rowA + k0 + 2 * kh);
        v2f b = *(const v2f*)(rowB + k0 + 2 * kh);
        // D = A(16x4) * B(4x16) + C, f32 in/out
        c = __builtin_amdgcn_wmma_f32_16x16x4_f32(
                /*neg_a=*/false, a, /*neg_b=*/false, b,
                /*c_mod=*/(short)0, c, /*reuse_a=*/false, /*reuse_b=*/false);
    }

    // Diagonal extraction (C/D layout: lanes 0..15 hold M=vgpr, lanes 16..31
    // hold M=vgpr+8; N = lane&15). (n,n) lives in lane n / vgpr n for n<8 and
    // lane n+16 / vgpr n-8 for n>=8.
    const bool active = (lane < 8) || (lane >= 24);
    if (active) {
        const int v = (lane < 8) ? lane : (lane - 24);
        const int n = (lane < 8) ? lane : (lane - 16);
        float d = c[0];
        d = (v == 1) ? c[1] : d;
        d = (v == 2) ? c[2] : d;
        d = (v == 3) ? c[3] : d;
        d = (v == 4) ? c[4] : d;
        d = (v == 5) ? c[5] : d;
        d = (v == 6) ? c[6] : d;
        d = (v == 7) ? c[7] : d;

        const uint32_t idx = (uint32_t)(tile + n);
        const float u  = hash_u01(idx * 0x9E3779B9u ^ (seed * 0x85EBCA6Bu + 0x165667B1u));
        const float g  = -logf(-logf(u));                 // gumbel(0,1)
        const float ys = 1.0f / (1.0f + expf(-(d + g)));  // sigmoid(logit+g)
        // straight-through forward value: y_hard * y_soft, +1e-7
        const float wv = ((ys > THRESH) ? ys : 0.0f) + 1e-7f;
        w_out[tile + n] = wv;
    }
}

// ---------------------------------------------------------------------------
// Kernel 2: weighted neighbor scatter-add (segment_sum).
// 16 threads per edge, one float4 (16B) per thread; hardware f32 atomics.
// ---------------------------------------------------------------------------
__global__ void conv_scatter_kernel(const float* __restrict__ X,
                                    const int*   __restrict__ e0,
                                    const int*   __restrict__ e1,
                                    const float* __restrict__ w,
                                    float*       __restrict__ out,
                                    int E)
{
    const long t  = (long)blockIdx.x * blockDim.x + threadIdx.x;
    const long e  = t >> 4;
    if (e >= E) return;
    const int c4  = (int)(t & 15) * 4;

    const int   dst_row = e0[e];          // segment id (scatter target)
    const int   src_row = e1[e];          // gathered neighbor
    const float wv      = w[e];

    const float4 v = *(const float4*)(X + (size_t)src_row * DIM + c4);
    float* o = out + (size_t)dst_row * DIM + c4;
    unsafeAtomicAdd(o + 0, v.x * wv);     // global_atomic_add_f32
    unsafeAtomicAdd(o + 1, v.y * wv);
    unsafeAtomicAdd(o + 2, v.z * wv);
    unsafeAtomicAdd(o + 3, v.w * wv);
}

// ---------------------------------------------------------------------------
// Kernel 3: h = normalize(tmp [+ target]); acc += h; optional h_out = h.
// 16 lanes per row (float4 each); sum-of-squares reduced with shfl_xor 1..8
// (stays inside each 16-lane half of the wave32).
// ---------------------------------------------------------------------------
__global__ void normalize_acc_kernel(const float* __restrict__ tmp,
                                     const float* __restrict__ target, // nullable
                                     float*       __restrict__ acc,
                                     float*       __restrict__ h_out,  // nullable
                                     int n_rows)
{
    const long t   = (long)blockIdx.x * blockDim.x + threadIdx.x;
    const long row = t >> 4;
    if (row >= n_rows) return;
    const int c4 = (int)(t & 15) * 4;
    const size_t base = (size_t)row * DIM + c4;

    float4 v = *(const float4*)(tmp + base);
    if (target) {
        const float4 tg = *(const float4*)(target + base);
        v.x += tg.x; v.y += tg.y; v.z += tg.z; v.w += tg.w;
    }
    float s = v.x * v.x + v.y * v.y + v.z * v.z + v.w * v.w;
    s += __shfl_xor(s, 1, 32);
    s += __shfl_xor(s, 2, 32);
    s += __shfl_xor(s, 4, 32);
    s += __shfl_xor(s, 8, 32);

    const float r = 1.0f / fmaxf(sqrtf(s), 1e-12f);
    float4 h; h.x = v.x * r; h.y = v.y * r; h.z = v.z * r; h.w = v.w * r;

    float4 a = *(const float4*)(acc + base);
    a.x += h.x; a.y += h.y; a.z += h.z; a.w += h.w;
    *(float4*)(acc + base) = a;
    if (h_out) *(float4*)(h_out + base) = h;
}

// ---------------------------------------------------------------------------
// Kernel 4: final = mean of 6 behavior embeddings.
// ---------------------------------------------------------------------------
__global__ void fuse_kernel(const float* __restrict__ a, const float* __restrict__ b,
                            const float* __restrict__ c, const float* __restrict__ d,
                            const float* __restrict__ e, const float* __restrict__ f,
                            float* __restrict__ out, long n4)
{
    const long t = (long)blockIdx.x * blockDim.x + threadIdx.x;
    if (t >= n4) return;
    const size_t i = (size_t)t * 4;
    const float4 va = *(const float4*)(a + i);
    const float4 vb = *(const float4*)(b + i);
    const float4 vc = *(const float4*)(c + i);
    const float4 vd = *(const float4*)(d + i);
    const float4 ve = *(const float4*)(e + i);
    const float4 vf = *(const float4*)(f + i);
    float4 o;
    const float k = 1.0f / 6.0f;
    o.x = (va.x + vb.x + vc.x + vd.x + ve.x + vf.x) * k;
    o.y = (va.y + vb.y + vc.y + vd.y + ve.y + vf.y) * k;
    o.z = (va.z + vb.z + vc.z + vd.z + ve.z + vf.z) * k;
    o.w = (va.w + vb.w + vc.w + vd.w + ve.w + vf.w) * k;
    *(float4*)(out + i) = o;
}

// ---------------------------------------------------------------------------
// Host orchestration (graph-capture safe: kernels + async memcpy/memset only).
// ---------------------------------------------------------------------------
extern "C" void kernel_launch(void* const* d_in, const int* in_sizes, int n_in,
                              void* d_out, int out_size, void* d_ws, size_t ws_size,
                              hipStream_t stream)
{
    const float* user_emb = (const float*)d_in[0];
    const float* item_emb = (const float*)d_in[1];
    const int* edge_ubg      = (const int*)d_in[2];
    const int* edge_view     = (const int*)d_in[3];
    const int* edge_cart     = (const int*)d_in[4];
    const int* edge_buy      = (const int*)d_in[5];
    const int* edge_view_buy = (const int*)d_in[6];
    const int* edge_cart_buy = (const int*)d_in[7];

    float* ws = (float*)d_ws;
    const size_t S = (size_t)NTOT * DIM;   // 12,800,128 floats per table
    float* X    = ws + 0 * S;   // init concat
    float* UBG  = ws + 1 * S;
    float* VIEW = ws + 2 * S;
    float* CART = ws + 3 * S;
    float* BUY  = ws + 4 * S;
    float* VB   = ws + 5 * S;
    float* CB   = ws + 6 * S;
    float* TMP  = ws + 7 * S;   // conv scratch
    float* H    = ws + 8 * S;   // layer state for 3-layer propagate
    float* W    = ws + 9 * S;   // edge weights [E]
    // requires ws_size >= (9*S + NEDGE)*4 bytes (~466 MB)

    // ---- concat [user_emb ; item_emb] -> X
    hipMemcpyAsync(X, user_emb, (size_t)in_sizes[0] * sizeof(float),
                   hipMemcpyDeviceToDevice, stream);
    hipMemcpyAsync(X + in_sizes[0], item_emb, (size_t)in_sizes[1] * sizeof(float),
                   hipMemcpyDeviceToDevice, stream);

    const int lrn_threads = (NEDGE / 16) * 32;               // 16 edges / wave
    const int lrn_blocks  = (lrn_threads + 255) / 256;
    const int sc_blocks   = (int)(((long)NEDGE * 16 + 255) / 256);
    const int nm_blocks   = (int)(((long)NTOT * 16 + 255) / 256);

#define LEARN(edge, table, seed)                                               \
    edge_gumbel_wmma_kernel<<<lrn_blocks, 256, 0, stream>>>(                   \
        table, edge, (edge) + NEDGE, W, NEDGE, (uint32_t)(seed))

#define CONV(xsrc, edge, target, accp, houtp)                                  \
    do {                                                                       \
        hipMemsetAsync(TMP, 0, S * sizeof(float), stream);                     \
        conv_scatter_kernel<<<sc_blocks, 256, 0, stream>>>(                    \
            xsrc, edge, (edge) + NEDGE, W, TMP, NEDGE);                        \
        normalize_acc_kernel<<<nm_blocks, 256, 0, stream>>>(                   \
            TMP, target, accp, houtp, NTOT);                                   \
    } while (0)

    // ---- ubg: 1 layer from X
    LEARN(edge_ubg, X, 1);
    hipMemcpyAsync(UBG, X, S * sizeof(float), hipMemcpyDeviceToDevice, stream);
    CONV(X, edge_ubg, (const float*)nullptr, UBG, (float*)nullptr);

    // ---- view / cart / buy: 1 layer each from UBG
    LEARN(edge_view, UBG, 2);
    hipMemcpyAsync(VIEW, UBG, S * sizeof(float), hipMemcpyDeviceToDevice, stream);
    CONV(UBG, edge_view, (const float*)nullptr, VIEW, (float*)nullptr);

    LEARN(edge_cart, UBG, 3);
    hipMemcpyAsync(CART, UBG, S * sizeof(float), hipMemcpyDeviceToDevice, stream);
    CONV(UBG, edge_cart, (const float*)nullptr, CART, (float*)nullptr);

    LEARN(edge_buy, UBG, 4);
    hipMemcpyAsync(BUY, UBG, S * sizeof(float), hipMemcpyDeviceToDevice, stream);
    CONV(UBG, edge_buy, (const float*)nullptr, BUY, (float*)nullptr);

    // ---- view_buy: 1 layer from VIEW
    LEARN(edge_view_buy, VIEW, 5);
    hipMemcpyAsync(VB, VIEW, S * sizeof(float), hipMemcpyDeviceToDevice, stream);
    CONV(VIEW, edge_view_buy, (const float*)nullptr, VB, (float*)nullptr);

    // ---- cart_buy: 3 layers from CART, target = BUY
    LEARN(edge_cart_buy, CART, 6);
    hipMemcpyAsync(CB, CART, S * sizeof(float), hipMemcpyDeviceToDevice, stream);
    hipMemcpyAsync(H,  CART, S * sizeof(float), hipMemcpyDeviceToDevice, stream);
    for (int l = 0; l < 3; ++l) {
        CONV(H, edge_cart_buy, BUY, CB, H);   // h = norm(conv(h)+buy); acc += h
    }

    // ---- fusion: mean of six behavior embeddings -> d_out
    const long n4 = (long)(S / 4);
    fuse_kernel<<<(int)((n4 + 255) / 256), 256, 0, stream>>>(
        UBG, VIEW, CART, BUY, VB, CB, (float*)d_out, n4);

#undef LEARN
#undef CONV
}